// L3LGConv_84859963834451
// MI455X (gfx1250) — compile-verified
//
#include <hip/hip_runtime.h>
#include <stdint.h>

// ---------------------------------------------------------------------------
// L3 LightGCN (LGConv x3 + relu) for MI455X / gfx1250.
//
// Bandwidth/atomic bound (AI ~0.25 FLOP/B) -> no WMMA applies. Uses the
// CDNA5 async gather path: per-lane global_load_async_to_lds_b128 pulls the
// 512B feature row of edge e+nwaves into LDS while the wave scatters edge e with
// global_atomic_add_f32. Tracked with ASYNCcnt (s_wait_asynccnt).
// ---------------------------------------------------------------------------

#define BLOCK 256
#define WAVES_PER_BLOCK 8   // wave32: 256 threads = 8 waves

// ---- CDNA5 async gather helpers (inline asm; portable across toolchains) --
__device__ __forceinline__ void async_gather_b128(unsigned lds_off,
                                                  const float* gaddr) {
  // vdst = LDS byte address (VGPR), vaddr = 64-bit global address (VGPR pair)
  asm volatile("global_load_async_to_lds_b128 %0, %1, off"
               :: "v"(lds_off), "v"(gaddr)
               : "memory");
}
__device__ __forceinline__ void wait_async_le1() {
  asm volatile("s_wait_asynccnt 0x1" ::: "memory");
}
__device__ __forceinline__ void wait_async_le0() {
  asm volatile("s_wait_asynccnt 0x0" ::: "memory");
}

// ---- degree: deg[dst] += 1 ------------------------------------------------
__global__ __launch_bounds__(BLOCK) void deg_kernel(
    const int* __restrict__ ei, float* __restrict__ deg, int E) {
  int e = blockIdx.x * BLOCK + threadIdx.x;
  if (e < E) atomicAdd(&deg[ei[(size_t)E + e]], 1.0f);
}

// ---- dinv = deg > 0 ? rsqrt(deg) : 0 -------------------------------------
__global__ __launch_bounds__(BLOCK) void dinv_kernel(
    const float* __restrict__ deg, float* __restrict__ dinv, int N) {
  int i = blockIdx.x * BLOCK + threadIdx.x;
  if (i < N) {
    float d = deg[i];
    dinv[i] = (d > 0.0f) ? rsqrtf(d) : 0.0f;
  }
}

// ---- one LGConv: xout[dst] += dinv[src]*dinv[dst]*f(xin[src]) -------------
// One wave per edge; each lane owns 16B (4 floats) of the 512B row.
// Depth-2 software pipeline: async-gather row(e+stride) to LDS while
// processing row(e). RELU_IN fuses the previous layer's relu into the read.
template <int RELU_IN>
__global__ __launch_bounds__(BLOCK) void spmm_scatter(
    const float* __restrict__ xin, float* __restrict__ xout,
    const int* __restrict__ ei, const float* __restrict__ dinv,
    int E, int nwaves) {
  __shared__ __align__(16) float smem[WAVES_PER_BLOCK * 2 * 128];
  const int lane = threadIdx.x & 31;
  const int wid  = threadIdx.x >> 5;
  const int gw   = blockIdx.x * WAVES_PER_BLOCK + wid;

  float* wbuf = &smem[wid * 256];  // two 128-float buffers for this wave
  const unsigned lds_base =
      (unsigned)(unsigned long long)wbuf + (unsigned)lane * 16u;

  int e = gw;
  int sCur = 0;
  int cur = 0;
  if (e < E) {  // prologue: gather row of first edge
    sCur = ei[e];
    async_gather_b128(lds_base, xin + (size_t)sCur * 128 + lane * 4);
  }
  while (e < E) {  // wave-uniform loop bounds: EXEC stays all-ones
    const int en = e + nwaves;
    int sN = 0;
    if (en < E) {
      sN = ei[en];
      async_gather_b128(lds_base + (unsigned)(cur ^ 1) * 512u,
                        xin + (size_t)sN * 128 + lane * 4);
      wait_async_le1();  // in-order: current buffer has landed
    } else {
      wait_async_le0();
    }
    const int d = ei[(size_t)E + e];
    const float nrm = dinv[sCur] * dinv[d];
    const float4 v = *(const float4*)(wbuf + cur * 128 + lane * 4);
    float4 m;
    if (RELU_IN) {
      m.x = fmaxf(v.x, 0.0f) * nrm;
      m.y = fmaxf(v.y, 0.0f) * nrm;
      m.z = fmaxf(v.z, 0.0f) * nrm;
      m.w = fmaxf(v.w, 0.0f) * nrm;
    } else {
      m.x = v.x * nrm; m.y = v.y * nrm; m.z = v.z * nrm; m.w = v.w * nrm;
    }
    float* o = xout + (size_t)d * 128 + lane * 4;
    atomicAdd(o + 0, m.x);  // global_atomic_add_f32, no-return (STOREcnt)
    atomicAdd(o + 1, m.y);
    atomicAdd(o + 2, m.z);
    atomicAdd(o + 3, m.w);
    sCur = sN;
    e = en;
    cur ^= 1;
  }
}

// ---- final relu: out = max(in, 0) -----------------------------------------
__global__ __launch_bounds__(BLOCK) void relu_kernel(
    const float4* __restrict__ in, float4* __restrict__ out, int n4) {
  int i = blockIdx.x * BLOCK + threadIdx.x;
  if (i < n4) {
    float4 v = in[i];
    v.x = fmaxf(v.x, 0.0f);
    v.y = fmaxf(v.y, 0.0f);
    v.z = fmaxf(v.z, 0.0f);
    v.w = fmaxf(v.w, 0.0f);
    out[i] = v;
  }
}

extern "C" void kernel_launch(void* const* d_in, const int* in_sizes, int n_in,
                              void* d_out, int out_size, void* d_ws,
                              size_t ws_size, hipStream_t stream) {
  (void)n_in; (void)out_size; (void)ws_size;
  const float* x  = (const float*)d_in[0];      // [N,128] f32
  const int*   ei = (const int*)d_in[1];        // [2,E] (integer -> int32 per harness)
  const int N = in_sizes[0] / 128;
  const int E = in_sizes[1] / 2;

  // Workspace layout: deg[N] | dinv[N] | A[N*128]   (~26 MB)
  float* deg  = (float*)d_ws;
  float* dinv = deg + N;
  float* A    = dinv + N;                       // 400000B offset: 16B aligned
  float* out  = (float*)d_out;

  const size_t featBytes = (size_t)N * 128 * sizeof(float);

  // degree + normalization
  hipMemsetAsync(deg, 0, (size_t)N * sizeof(float), stream);
  deg_kernel<<<(E + BLOCK - 1) / BLOCK, BLOCK, 0, stream>>>(ei, deg, E);
  dinv_kernel<<<(N + BLOCK - 1) / BLOCK, BLOCK, 0, stream>>>(deg, dinv, N);

  const int SP_BLOCKS = 4096;
  const int nwaves = SP_BLOCKS * WAVES_PER_BLOCK;

  // conv1: x -> A
  hipMemsetAsync(A, 0, featBytes, stream);
  spmm_scatter<0><<<SP_BLOCKS, BLOCK, 0, stream>>>(x, A, ei, dinv, E, nwaves);
  // conv2 (relu fused on read): A -> out
  hipMemsetAsync(out, 0, featBytes, stream);
  spmm_scatter<1><<<SP_BLOCKS, BLOCK, 0, stream>>>(A, out, ei, dinv, E, nwaves);
  // conv3 (relu fused on read): out -> A
  hipMemsetAsync(A, 0, featBytes, stream);
  spmm_scatter<1><<<SP_BLOCKS, BLOCK, 0, stream>>>(out, A, ei, dinv, E, nwaves);
  // final relu: out = relu(A)
  const int n4 = N * 128 / 4;
  relu_kernel<<<(n4 + BLOCK - 1) / BLOCK, BLOCK, 0, stream>>>(
      (const float4*)A, (float4*)out, n4);
}